// StokenAttention_26938034880603
// MI455X (gfx1250) — compile-verified
//
#include <hip/hip_runtime.h>
#include <math.h>

#define B_     8
#define C_     192
#define H_     224
#define W_     224
#define SH_    28
#define SS_    784     // 28*28
#define N_     784
#define K3_    9
#define HEADS_ 8
#define HD_    24
#define OC_    576     // 3*C

typedef __attribute__((ext_vector_type(2))) float v2f;
typedef __attribute__((ext_vector_type(8))) float v8f;

#if defined(__has_builtin)
#if __has_builtin(__builtin_amdgcn_tensor_load_to_lds) && \
    __has_builtin(__builtin_amdgcn_s_wait_tensorcnt)
#define HAVE_TDM 1
#endif
#endif

#ifdef HAVE_TDM
typedef __attribute__((ext_vector_type(4))) unsigned int u32x4;
typedef __attribute__((ext_vector_type(8))) int          i32x8;
typedef __attribute__((ext_vector_type(4))) int          i32x4;
#endif

// ---- WMMA f32 16x16x4 helpers (wave32, CDNA5 VGPR layouts) ------------------
// A 16x4 (MxK): lane<16 -> M=lane, K={0,1}; lane>=16 -> M=lane-16, K={2,3}
// B 4x16 (KxN): vgpr v, lane<16 -> K=v, N=lane; lane>=16 -> K=v+2, N=lane-16
// Both reduce to: r = lane&15, kb = 2*(lane>>4); elem(r, kb+i)
__device__ __forceinline__ v2f load_frag(const float* p, int ld_r, int ld_k) {
  const int lane = threadIdx.x & 31;
  const int r  = lane & 15;
  const int kb = (lane >> 4) << 1;
  v2f f;
  f.x = p[r * ld_r +  kb      * ld_k];
  f.y = p[r * ld_r + (kb + 1) * ld_k];
  return f;
}

__device__ __forceinline__ v8f wmma4(v2f a, v2f b, v8f c) {
  return __builtin_amdgcn_wmma_f32_16x16x4_f32(false, a, false, b, (short)0, c,
                                               false, false);
}

// C/D 16x16 f32: vgpr i, lane<16 -> (M=i, N=lane); lane>=16 -> (M=i+8, N=lane-16)
__device__ __forceinline__ void store_c(float* p, v8f c, int ld_m, int ld_n,
                                        int mmax, int nmax) {
  const int lane = threadIdx.x & 31;
  const int n    = lane & 15;
  const int mofs = (lane >> 4) << 3;
  if (n < nmax) {
#pragma unroll
    for (int i = 0; i < 8; ++i) {
      const int m = mofs + i;
      if (m < mmax) p[m * ld_m + n * ld_n] = c[i];
    }
  }
}

// ---- 1: super-token means ---------------------------------------------------
__global__ void k_stoken_mean(const float* __restrict__ x,
                              float* __restrict__ stoken) {
  int idx = blockIdx.x * blockDim.x + threadIdx.x;
  if (idx >= B_ * C_ * SS_) return;
  const int sx = idx % SH_; int t = idx / SH_;
  const int sy = t % SH_;   t /= SH_;
  const int c  = t % C_;    const int b = t / C_;
  const float* base = x + (((size_t)(b * C_ + c) * H_) + sy * 8) * W_ + sx * 8;
  __builtin_prefetch(base + 4 * W_, 0, 1);
  float s = 0.f;
#pragma unroll
  for (int i = 0; i < 8; ++i)
#pragma unroll
    for (int j = 0; j < 8; ++j) s += base[i * W_ + j];
  stoken[idx] = s * (1.0f / 64.0f);
}

// ---- 2: affinity + softmax + per-block super-token update -------------------
#define PIXLD 196
__global__ void k_affinity(const float* __restrict__ x,
                           const float* __restrict__ stoken,
                           float* __restrict__ affG,
                           float* __restrict__ sG,
                           float* __restrict__ stblkG) {
  extern __shared__ float smem[];
  float* sPix = smem;                  // 64 x PIXLD  (pixel block, c fastest)
  float* sSf  = sPix + 64 * PIXLD;     // 16 x 192    (rows 9..15 zero)
  float* sAff = sSf + 16 * C_;         // 64 x 16
  const int bn = blockIdx.x;
  const int b  = bn / N_;
  const int n  = bn % N_;
  const int y0 = n / SH_, x0 = n % SH_;
  const int t  = threadIdx.x;
  const int wave = t >> 5;

  for (int u = t; u < C_ * 8; u += 256) {           // 64x192 pixel tile
    const int c = u >> 3, i = u & 7;
    const float* gp = x + (((size_t)(b * C_ + c) * H_) + (y0 * 8 + i)) * W_ + x0 * 8;
#pragma unroll
    for (int j = 0; j < 8; ++j) sPix[(i * 8 + j) * PIXLD + c] = gp[j];
  }
  for (int e = t; e < 16 * C_; e += 256) {          // 3x3 stoken neighborhood
    const int kk = e / C_, c = e % C_;
    float v = 0.f;
    if (kk < 9) {
      const int sy = y0 + kk / 3 - 1, sx = x0 + kk % 3 - 1;
      if (sy >= 0 && sy < SH_ && sx >= 0 && sx < SH_)
        v = stoken[((size_t)(b * C_ + c) * SH_ + sy) * SH_ + sx];
    }
    sSf[e] = v;
  }
  __syncthreads();

  const float scale = rsqrtf((float)C_);
  // aff(64x9) = pixel(64x192) x sf^T(192x9)   (N padded to 16)
  if (wave < 4) {
    v8f acc = {};
    for (int c0 = 0; c0 < C_; c0 += 4) {
      v2f a  = load_frag(&sPix[(wave * 16) * PIXLD + c0], PIXLD, 1);
      v2f bf = load_frag(&sSf[c0], C_, 1);
      acc = wmma4(a, bf, acc);
    }
    acc = acc * scale;
    store_c(&sAff[(wave * 16) * 16], acc, 16, 1, 16, 16);
  }
  __syncthreads();

  if (t < 64) {                                     // softmax over 9 neighbors
    const int p = t;
    float m = -3.0e38f;
#pragma unroll
    for (int k = 0; k < 9; ++k) m = fmaxf(m, sAff[p * 16 + k]);
    float e[9], s = 0.f;
#pragma unroll
    for (int k = 0; k < 9; ++k) { e[k] = __expf(sAff[p * 16 + k] - m); s += e[k]; }
    const float inv = 1.0f / s;
    float* gout = affG + ((size_t)bn * 64 + p) * 9;
#pragma unroll
    for (int k = 0; k < 9; ++k) {
      const float v = e[k] * inv;
      sAff[p * 16 + k] = v;
      gout[k] = v;
    }
  }
  __syncthreads();
  if (t < 9) {
    float s = 0.f;
    for (int p = 0; p < 64; ++p) s += sAff[p * 16 + t];
    sG[(size_t)bn * 9 + t] = s;
  }
  __syncthreads();

  // st_blk(192x9) = pixel^T(192x64) x aff(64x9)
  for (int ct = wave; ct < 12; ct += 8) {
    v8f acc = {};
    for (int p0 = 0; p0 < 64; p0 += 4) {
      v2f a  = load_frag(&sPix[p0 * PIXLD + ct * 16], 1, PIXLD);
      v2f bf = load_frag(&sAff[p0 * 16], 1, 16);
      acc = wmma4(a, bf, acc);
    }
    store_c(stblkG + ((size_t)bn * C_ + ct * 16) * 9, acc, 9, 1, 16, 9);
  }
}

// ---- 3/4: fold3 adjoint gathers --------------------------------------------
__global__ void k_amsum(const float* __restrict__ sG, float* __restrict__ am) {
  int idx = blockIdx.x * blockDim.x + threadIdx.x;
  if (idx >= B_ * SS_) return;
  const int xx = idx % SH_; int t = idx / SH_;
  const int yy = t % SH_;  const int b = t / SH_;
  float acc = 0.f;
#pragma unroll
  for (int di = 0; di < 3; ++di)
#pragma unroll
    for (int dj = 0; dj < 3; ++dj) {
      const int ny = yy + 1 - di, nx = xx + 1 - dj;
      if (ny >= 0 && ny < SH_ && nx >= 0 && nx < SH_)
        acc += sG[((size_t)b * N_ + ny * SH_ + nx) * 9 + di * 3 + dj];
    }
  am[idx] = acc;
}

__global__ void k_fold(const float* __restrict__ stblkG,
                       const float* __restrict__ am,
                       float* __restrict__ stf) {
  int idx = blockIdx.x * blockDim.x + threadIdx.x;
  if (idx >= B_ * C_ * SS_) return;
  const int xx = idx % SH_; int t = idx / SH_;
  const int yy = t % SH_;  t /= SH_;
  const int c = t % C_;    const int b = t / C_;
  float acc = 0.f;
#pragma unroll
  for (int di = 0; di < 3; ++di)
#pragma unroll
    for (int dj = 0; dj < 3; ++dj) {
      const int ny = yy + 1 - di, nx = xx + 1 - dj;
      if (ny >= 0 && ny < SH_ && nx >= 0 && nx < SH_)
        acc += stblkG[(((size_t)b * N_ + ny * SH_ + nx) * C_ + c) * 9 + di * 3 + dj];
    }
  const float d = am[b * SS_ + yy * SH_ + xx] + 1e-12f;
  stf[((size_t)(b * C_ + c)) * SS_ + yy * SH_ + xx] = acc / d;
}

// ---- 5/7: channel GEMM  D(b) = W(Mx192) * src(b)(192x784) (+bias) ----------
__global__ void k_gemm_chan(const float* __restrict__ Wmat,
                            const float* __restrict__ src,
                            const float* __restrict__ bias,
                            float* __restrict__ dst, int M) {
  const int wave = threadIdx.x >> 5;
  const int MT = M >> 4;
  const int tiles_per_b = MT * 49;
  const int tile = blockIdx.x * 8 + wave;
  if (tile >= B_ * tiles_per_b) return;
  const int b = tile / tiles_per_b;
  const int r = tile % tiles_per_b;
  const int mt = r / 49, nt = r % 49;
  const float* srcb = src + (size_t)b * C_ * SS_;
  v8f acc = {};
  for (int c0 = 0; c0 < C_; c0 += 4) {
    v2f a  = load_frag(Wmat + (size_t)(mt * 16) * C_ + c0, C_, 1);
    v2f bf = load_frag(srcb + (size_t)c0 * SS_ + nt * 16, 1, SS_);
    acc = wmma4(a, bf, acc);
  }
  float* dbase = dst + ((size_t)b * M + mt * 16) * SS_ + nt * 16;
  if (bias) {
    const int lane = threadIdx.x & 31;
    const int n = lane & 15;
    const int mofs = (lane >> 4) << 3;
#pragma unroll
    for (int i = 0; i < 8; ++i) {
      const int m = mofs + i;
      dbase[m * SS_ + n] = acc[i] + bias[mt * 16 + m];
    }
  } else {
    store_c(dbase, acc, SS_, 1, 16, 16);
  }
}

// ---- 6: multi-head attention over 784 tokens (K,V,Q resident in LDS) -------
__global__ void k_attn(const float* __restrict__ qkvG, float* __restrict__ ao) {
  extern __shared__ float smem[];
  float* Qs  = smem;                   // 24 x 784
  float* Ks  = Qs + HD_ * SS_;         // 24 x 784
  float* Vs  = Ks + HD_ * SS_;         // 24 x 784
  float* Sb  = Vs + HD_ * SS_;         // 784 x 16
  float* red = Sb + SS_ * 16;          // 256
  float* gm  = red + 256;              // 16
  float* gs  = gm + 16;                // 16
  const int b = blockIdx.x / HEADS_;
  const int h = blockIdx.x % HEADS_;
  const int t = threadIdx.x;
  const int wave = t >> 5;
  const float scale = rsqrtf((float)HD_);

  // per-head qkv rows h*72 .. h*72+71 are q(24), k(24), v(24) in order
  const float* qkvb = qkvG + ((size_t)b * OC_ + h * 72) * SS_;
#ifdef HAVE_TDM
  // Tensor Data Mover: DMA the contiguous 72x784-float Q/K/V slab into LDS.
  // 1-D tile: tile_dim0 = tensor_dim0 = 72*784 = 56448 elements of 4 bytes.
  if (wave == 0) {
    const unsigned lds0  = (unsigned)(unsigned long long)(const void*)smem;
    const unsigned long long ga = (unsigned long long)(const void*)qkvb;
    const unsigned nelem = 72u * SS_;            // 56448 (< 2^16)
    u32x4 g0;
    g0[0] = 1u;                                  // count=1, user descriptor
    g0[1] = lds0;                                // lds_addr (bytes)
    g0[2] = (unsigned)ga;                        // global_addr[31:0]
    g0[3] = (unsigned)((ga >> 32) & 0x01FFFFFFu) | (2u << 30); // addr[56:32] | type=2
    i32x8 g1;
    g1[0] = (int)(2u << 16);                     // data_size=4B; no mask/barrier/pad
    g1[1] = (int)((nelem & 0xFFFFu) << 16);      // tensor_dim0[15:0] in [63:48]
    g1[2] = (int)(1u << 16);                     // tensor_dim0 hi=0, tensor_dim1=1
    g1[3] = (int)((nelem & 0xFFFFu) << 16);      // tile_dim0 in [127:112]
    g1[4] = 1;                                   // tile_dim1=1, tile_dim2=0
    g1[5] = (int)nelem;                          // tensor_dim0_stride lo32
    g1[6] = 0;                                   // stride hi, tensor_dim1_stride lo
    g1[7] = 0;
    i32x4 gz4 = {0, 0, 0, 0};
    i32x8 gz8 = {0, 0, 0, 0, 0, 0, 0, 0};
    __builtin_amdgcn_tensor_load_to_lds(g0, g1, gz4, gz4, gz8, 0);
    __builtin_amdgcn_s_wait_tensorcnt(0);
  }
#else
  for (int e = t; e < 72 * SS_; e += 256) smem[e] = qkvb[e];
#endif
  __syncthreads();

  for (int mt = 0; mt < 49; ++mt) {
    for (int nt = wave; nt < 49; nt += 8) {        // S = K^T x Q-tile
      v8f acc = {};
#pragma unroll
      for (int d0 = 0; d0 < HD_; d0 += 4) {
        v2f a  = load_frag(&Ks[d0 * SS_ + nt * 16], 1, SS_);
        v2f bf = load_frag(&Qs[d0 * SS_ + mt * 16], 1, SS_);
        acc = wmma4(a, bf, acc);
      }
      acc = acc * scale;
      store_c(&Sb[(nt * 16) * 16], acc, 16, 1, 16, 16);
    }
    __syncthreads();

    // stable softmax over 784 keys, per query column
    const int q = t & 15, part = t >> 4;
    const int nlo = part * 49;
    float lm = -3.0e38f;
    for (int i = 0; i < 49; ++i) lm = fmaxf(lm, Sb[(nlo + i) * 16 + q]);
    red[t] = lm;
    __syncthreads();
    if (t < 16) {
      float g = -3.0e38f;
      for (int j = 0; j < 16; ++j) g = fmaxf(g, red[j * 16 + t]);
      gm[t] = g;
    }
    __syncthreads();
    float ls = 0.f;
    for (int i = 0; i < 49; ++i) {
      const float e = __expf(Sb[(nlo + i) * 16 + q] - gm[q]);
      Sb[(nlo + i) * 16 + q] = e;
      ls += e;
    }
    red[t] = ls;
    __syncthreads();
    if (t < 16) {
      float g = 0.f;
      for (int j = 0; j < 16; ++j) g += red[j * 16 + t];
      gs[t] = g;
    }
    __syncthreads();
    const float inv = 1.0f / gs[q];
    for (int i = 0; i < 49; ++i) Sb[(nlo + i) * 16 + q] *= inv;
    __syncthreads();

    if (wave < 2) {                                // out = V x P
      v8f acc = {};
      for (int n0 = 0; n0 < SS_; n0 += 4) {
        v2f a  = load_frag(&Vs[(wave * 16) * SS_ + n0], SS_, 1);
        v2f bf = load_frag(&Sb[n0 * 16], 1, 16);
        acc = wmma4(a, bf, acc);
      }
      float* dbase = ao + ((size_t)b * C_ + h * HD_ + wave * 16) * SS_ + mt * 16;
      store_c(dbase, acc, SS_, 1, wave == 0 ? 16 : 8, 16);
    }
    __syncthreads();
  }
}

// ---- 8: broadcast refined tokens back to pixels ----------------------------
__global__ void k_pix(const float* __restrict__ stref,
                      const float* __restrict__ affG,
                      float* __restrict__ out) {
  extern __shared__ float smem[];
  float* sSf = smem;              // 12 x 192  (rows 9..11 zero)
  float* sA  = sSf + 12 * C_;     // 64 x 12   (cols 9..11 zero)
  const int bn = blockIdx.x;
  const int b = bn / N_, n = bn % N_;
  const int y0 = n / SH_, x0 = n % SH_;
  const int t = threadIdx.x;
  const int wave = t >> 5;

  for (int e = t; e < 12 * C_; e += 256) {
    const int kk = e / C_, c = e % C_;
    float v = 0.f;
    if (kk < 9) {
      const int sy = y0 + kk / 3 - 1, sx = x0 + kk % 3 - 1;
      if (sy >= 0 && sy < SH_ && sx >= 0 && sx < SH_)
        v = stref[((size_t)(b * C_ + c)) * SS_ + sy * SH_ + sx];
    }
    sSf[e] = v;
  }
  for (int e = t; e < 64 * 12; e += 256) {
    const int p = e / 12, kk = e % 12;
    sA[e] = (kk < 9) ? affG[((size_t)bn * 64 + p) * 9 + kk] : 0.f;
  }
  __syncthreads();

  // pix(192x64) = sf(192x12pad) x aff^T(12x64)
  for (int tt = wave; tt < 48; tt += 8) {
    const int ct = tt % 12, pt = tt / 12;
    v8f acc = {};
#pragma unroll
    for (int k0 = 0; k0 < 12; k0 += 4) {
      v2f a  = load_frag(&sSf[k0 * C_ + ct * 16], 1, C_);
      v2f bf = load_frag(&sA[(pt * 16) * 12 + k0], 12, 1);
      acc = wmma4(a, bf, acc);
    }
    const int lane = t & 31;
    const int nn = lane & 15;
    const int mofs = (lane >> 4) << 3;
    const int p = pt * 16 + nn;
    const int py = p >> 3, px = p & 7;
    float* ob = out + (size_t)b * C_ * H_ * W_ +
                (size_t)(y0 * 8 + py) * W_ + (x0 * 8 + px);
#pragma unroll
    for (int i = 0; i < 8; ++i) {
      const int c = ct * 16 + mofs + i;
      ob[(size_t)c * H_ * W_] = acc[i];
    }
  }
}

// ---- host ------------------------------------------------------------------
extern "C" void kernel_launch(void* const* d_in, const int* in_sizes, int n_in,
                              void* d_out, int out_size, void* d_ws, size_t ws_size,
                              hipStream_t stream) {
  (void)in_sizes; (void)n_in; (void)out_size; (void)ws_size;
  const float* x     = (const float*)d_in[0];
  const float* Wqkv  = (const float*)d_in[1];
  const float* Wproj = (const float*)d_in[2];
  const float* bproj = (const float*)d_in[3];
  float* out = (float*)d_out;
  float* ws  = (float*)d_ws;

  size_t o = 0;
  float* stoken = ws + o; o += (size_t)B_ * C_ * SS_;        // reused as st_refined
  float* aff    = ws + o; o += (size_t)B_ * N_ * 64 * K3_;
  float* ssum   = ws + o; o += (size_t)B_ * N_ * K3_;
  float* stblk  = ws + o; o += (size_t)B_ * N_ * C_ * K3_;
  float* am     = ws + o; o += (size_t)B_ * SS_;
  float* stf    = ws + o; o += (size_t)B_ * C_ * SS_;
  float* qkv    = ws + o; o += (size_t)B_ * OC_ * SS_;
  float* ao     = ws + o; o += (size_t)B_ * C_ * SS_;

  const int bcss_blocks = (B_ * C_ * SS_ + 255) / 256;        // 4704
  k_stoken_mean<<<bcss_blocks, 256, 0, stream>>>(x, stoken);

  const size_t aff_lds = (size_t)(64 * PIXLD + 16 * C_ + 64 * 16) * 4;
  k_affinity<<<B_ * N_, 256, aff_lds, stream>>>(x, stoken, aff, ssum, stblk);

  k_amsum<<<(B_ * SS_ + 255) / 256, 256, 0, stream>>>(ssum, am);
  k_fold<<<bcss_blocks, 256, 0, stream>>>(stblk, am, stf);

  const int qkv_tiles = B_ * (OC_ / 16) * 49;                 // 14112
  k_gemm_chan<<<(qkv_tiles + 7) / 8, 256, 0, stream>>>(Wqkv, stf, nullptr, qkv, OC_);

  const size_t attn_lds = (size_t)(3 * HD_ * SS_ + SS_ * 16 + 256 + 32) * 4;
  k_attn<<<B_ * HEADS_, 256, attn_lds, stream>>>(qkv, ao);

  float* stref = stoken;                                      // reuse
  const int proj_tiles = B_ * (C_ / 16) * 49;                 // 4704
  k_gemm_chan<<<(proj_tiles + 7) / 8, 256, 0, stream>>>(Wproj, ao, bproj, stref, C_);

  const size_t pix_lds = (size_t)(12 * C_ + 64 * 12) * 4;
  k_pix<<<B_ * N_, 256, pix_lds, stream>>>(stref, aff, out);
}